// SLAMFormer_23295902614135
// MI455X (gfx1250) — compile-verified
//
#include <hip/hip_runtime.h>
#include <hip/hip_bf16.h>

typedef __attribute__((ext_vector_type(2))) float v2f;
typedef __attribute__((ext_vector_type(8))) float v8f;

#define L_DIM 4096
#define C_DIM 1024
#define NEG_INF (-__builtin_inff())

// ---------------------------------------------------------------------------
// Kernel 1: row L2-normalize.  xn[row] = x[row] / max(||x[row]||, 1e-6)
// ---------------------------------------------------------------------------
__global__ void slam_normalize_rows(const float* __restrict__ x,
                                    float* __restrict__ xn) {
    __shared__ float red[256];
    const int row = blockIdx.x;
    const float* xr = x + (size_t)row * C_DIM;
    float s = 0.0f;
    for (int j = threadIdx.x; j < C_DIM; j += 256) {
        float v = xr[j];
        s += v * v;
    }
    red[threadIdx.x] = s;
    __syncthreads();
    for (int off = 128; off > 0; off >>= 1) {
        if (threadIdx.x < off) red[threadIdx.x] += red[threadIdx.x + off];
        __syncthreads();
    }
    float norm = fmaxf(sqrtf(red[0]), 1e-6f);
    float rinv = 1.0f / norm;
    float* xnr = xn + (size_t)row * C_DIM;
    for (int j = threadIdx.x; j < C_DIM; j += 256) xnr[j] = xr[j] * rinv;
}

// ---------------------------------------------------------------------------
// Kernel 2: cosine = 1 - xn @ xn^T via V_WMMA_F32_16X16X4_F32.
// Register-blocked: each wave computes a 2x2 grid of 16x16 tiles (32x32).
// Per K-step of 4: 4 fragment loads feed 4 WMMAs (A shared across the two
// N-tiles, B shared across the two M-tiles) -> 8 FLOP per L0/L2 byte.
// Block = 8 waves (2x4) -> 64x128 tile; grid 32x64.
//
// A (16x4 f32): lanes 0-15 hold M=lane, K={0,1}; lanes 16-31 M=lane-16,
// K={2,3}  -> per-lane float2 at xn[m*C + k0 + 2*(lane>>4)].
// B (4x16 f32): lanes 0-15 hold K={0,1} of column n=lane; lanes 16-31
// K={2,3}  -> per-lane float2 at xn[n*C + k0 + 2*(lane>>4)] (B = xn^T).
// C/D (16x16 f32): vgpr v, lanes 0-15 -> M=v, lanes 16-31 -> M=v+8, N=lane&15.
// ---------------------------------------------------------------------------
__global__ void slam_cosine_wmma(const float* __restrict__ xn,
                                 float* __restrict__ cosine) {
    const int lane = threadIdx.x & 31;
    const int wave = threadIdx.x >> 5;
    const int m_base = blockIdx.y * 64 + (wave >> 2) * 32;
    const int n_base = blockIdx.x * 128 + (wave & 3) * 32;
    const int half = lane >> 4;   // 0 or 1
    const int lm   = lane & 15;
    const int koff = half * 2;

    const float* pa0 = xn + (size_t)(m_base + lm) * C_DIM + koff;
    const float* pa1 = pa0 + (size_t)16 * C_DIM;
    const float* pb0 = xn + (size_t)(n_base + lm) * C_DIM + koff;
    const float* pb1 = pb0 + (size_t)16 * C_DIM;

    v8f c00 = {}, c01 = {}, c10 = {}, c11 = {};
#pragma unroll 4
    for (int k0 = 0; k0 < C_DIM; k0 += 4) {
        v2f a0 = *(const v2f*)(pa0 + k0);
        v2f a1 = *(const v2f*)(pa1 + k0);
        v2f b0 = *(const v2f*)(pb0 + k0);
        v2f b1 = *(const v2f*)(pb1 + k0);
        // 8 args: (neg_a, A, neg_b, B, c_mod, C, reuse_a, reuse_b)
        c00 = __builtin_amdgcn_wmma_f32_16x16x4_f32(false, a0, false, b0,
                                                    (short)0, c00, false, false);
        c01 = __builtin_amdgcn_wmma_f32_16x16x4_f32(false, a0, false, b1,
                                                    (short)0, c01, false, false);
        c10 = __builtin_amdgcn_wmma_f32_16x16x4_f32(false, a1, false, b0,
                                                    (short)0, c10, false, false);
        c11 = __builtin_amdgcn_wmma_f32_16x16x4_f32(false, a1, false, b1,
                                                    (short)0, c11, false, false);
    }

    const int n0 = n_base + lm;
#pragma unroll
    for (int v = 0; v < 8; ++v) {
        int m0 = m_base + v + 8 * half;
        cosine[(size_t)m0 * L_DIM + n0]             = 1.0f - c00[v];
        cosine[(size_t)m0 * L_DIM + n0 + 16]        = 1.0f - c01[v];
        cosine[(size_t)(m0 + 16) * L_DIM + n0]      = 1.0f - c10[v];
        cosine[(size_t)(m0 + 16) * L_DIM + n0 + 16] = 1.0f - c11[v];
    }
}

// ---------------------------------------------------------------------------
// Kernel 3: per-row two smallest values of cosine (cosine is symmetric, so
// rows == columns); initial_scores[row] = second smallest.
// ---------------------------------------------------------------------------
__global__ void slam_row_second_min(const float* __restrict__ cosine,
                                    float* __restrict__ init_scores) {
    __shared__ float s1[256];
    __shared__ float s2[256];
    const int row = blockIdx.x;
    const float* cr = cosine + (size_t)row * L_DIM;
    float a1 = __builtin_inff(), a2 = __builtin_inff();
    for (int j = threadIdx.x; j < L_DIM; j += 256) {
        float v = cr[j];
        if (v < a1) { a2 = a1; a1 = v; }
        else if (v < a2) { a2 = v; }
    }
    s1[threadIdx.x] = a1;
    s2[threadIdx.x] = a2;
    __syncthreads();
    for (int off = 128; off > 0; off >>= 1) {
        if (threadIdx.x < off) {
            float b1 = s1[threadIdx.x + off];
            float b2 = s2[threadIdx.x + off];
            float m1 = fminf(a1, b1);
            float m2 = fminf(fmaxf(a1, b1), fminf(a2, b2));
            a1 = m1; a2 = m2;
            s1[threadIdx.x] = a1;
            s2[threadIdx.x] = a2;
        }
        __syncthreads();
    }
    if (threadIdx.x == 0) init_scores[row] = s2[0];
}

// ---------------------------------------------------------------------------
// Kernel 4: greedy k-center selection, single 1024-thread workgroup.
// scores live in REGISTERS (4 per thread, contiguous ownership).  Each round:
//   per-thread 4-way scan -> wave32 shfl_xor butterfly (5 steps, no barrier)
//   -> 32 wave winners in LDS -> barrier -> wave 0 butterflies winners,
//   broadcasts argmax -> barrier -> coalesced b128 row read + register
//   min-update.  2 barriers per round (vs 11 for an LDS tree), which is the
//   serial-latency floor of the dependency chain.
// Tie-break: smaller index on equal value, matching jnp.argmax.
// ---------------------------------------------------------------------------
__global__ void slam_greedy_select(const float* __restrict__ cosine,
                                   const float* __restrict__ init_scores,
                                   const int* __restrict__ thr,
                                   float* __restrict__ out_sel) {
    __shared__ float redV[32];
    __shared__ int   redI[32];
    __shared__ int   bcast;
    __shared__ int   scanbuf[1024];

    const int tid  = threadIdx.x;
    const int lane = tid & 31;
    const int wid  = tid >> 5;
    const int j0   = tid * 4;

    int k = thr[0];
    if (k > L_DIM) k = L_DIM;
    if (k < 1) k = 1;

    float s[4];
#pragma unroll
    for (int u = 0; u < 4; ++u) s[u] = init_scores[j0 + u];
    unsigned fl = 0;   // per-thread selection flags (4 bits)

    for (int t = 0; t < k; ++t) {
        // ---- per-thread argmax over its 4 scores (earliest index wins) ----
        float bv = NEG_INF;
        int   bi = j0;
#pragma unroll
        for (int u = 0; u < 4; ++u) {
            if (s[u] > bv) { bv = s[u]; bi = j0 + u; }
        }
        // ---- wave32 butterfly reduction ----
#pragma unroll
        for (int off = 16; off > 0; off >>= 1) {
            float ov = __shfl_xor(bv, off, 32);
            int   oi = __shfl_xor(bi, off, 32);
            if (ov > bv || (ov == bv && oi < bi)) { bv = ov; bi = oi; }
        }
        if (lane == 0) { redV[wid] = bv; redI[wid] = bi; }
        __syncthreads();
        // ---- wave 0 reduces the 32 wave winners ----
        if (wid == 0) {
            bv = redV[lane];
            bi = redI[lane];
#pragma unroll
            for (int off = 16; off > 0; off >>= 1) {
                float ov = __shfl_xor(bv, off, 32);
                int   oi = __shfl_xor(bi, off, 32);
                if (ov > bv || (ov == bv && oi < bi)) { bv = ov; bi = oi; }
            }
            if (lane == 0) bcast = bi;
        }
        __syncthreads();
        const int idx = bcast;

        // ---- update: s = min(s, cosine[idx]); s[idx] = -inf ----
        // First round replaces scores entirely with cosine[first], matching
        // the reference's scores0 = cosine[first].
        const float* cr = cosine + (size_t)idx * L_DIM;
#pragma unroll
        for (int u = 0; u < 4; ++u) {
            int j = j0 + u;
            float c = cr[j];
            float v = (t == 0) ? c : fminf(s[u], c);
            if (j == idx) { v = NEG_INF; fl |= (1u << u); }
            s[u] = v;
        }
        // No barrier needed: the next round's shared-memory writes are
        // ordered after this round's reads by the two barriers above.
    }

    // ---- emit sorted selected indices via prefix scan of flag counts ----
    int cnt = __builtin_popcount(fl);
    scanbuf[tid] = cnt;
    __syncthreads();
    for (int off = 1; off < 1024; off <<= 1) {
        int v   = scanbuf[tid];
        int add = (tid >= off) ? scanbuf[tid - off] : 0;
        __syncthreads();
        scanbuf[tid] = v + add;
        __syncthreads();
    }
    int base = scanbuf[tid] - cnt;   // exclusive prefix
#pragma unroll
    for (int u = 0; u < 4; ++u) {
        if (fl & (1u << u)) out_sel[base++] = (float)(j0 + u);
    }
}

// ---------------------------------------------------------------------------
// Host-side launcher.
// d_in[0] = visual_feature_vectors (f32, 4096x1024)
// d_in[1] = image_feature_length   (int, scalar)  -- value fixed at 4096
// d_in[2] = threshold_terms        (int, scalar)
// d_out   = [ sel (1024 floats) | cosine (4096x4096 floats) ]
// d_ws    = [ xn (16 MB) | init_scores (16 KB) ]
// ---------------------------------------------------------------------------
extern "C" void kernel_launch(void* const* d_in, const int* in_sizes, int n_in,
                              void* d_out, int out_size, void* d_ws, size_t ws_size,
                              hipStream_t stream) {
    const float* x   = (const float*)d_in[0];
    const int*   thr = (const int*)d_in[2];

    float* out    = (float*)d_out;
    float* cosine = out + 1024;            // sel occupies first k=1024 slots
    float* xn     = (float*)d_ws;
    float* iscore = xn + (size_t)L_DIM * C_DIM;

    slam_normalize_rows<<<L_DIM, 256, 0, stream>>>(x, xn);

    dim3 ggrid(L_DIM / 128, L_DIM / 64);
    slam_cosine_wmma<<<ggrid, 256, 0, stream>>>(xn, cosine);

    slam_row_second_min<<<L_DIM, 256, 0, stream>>>(cosine, iscore);

    slam_greedy_select<<<1, 1024, 0, stream>>>(cosine, iscore, thr, out);
}